// MyMSE_7000796692659
// MI455X (gfx1250) — compile-verified
//
#include <hip/hip_runtime.h>
#include <stdint.h>
#include <stddef.h>

// MyMSE segmented reduction for MI455X (gfx1250, wave32).
// global -> LDS via 4-deep global_load_async_to_lds_b128 pipeline (ASYNCcnt,
// th:TH_LOAD_NT since 256MB streams once through 192MB L2), per-lane
// bank-conflict-free LDS bins via ds_add_f32, deterministic 3-level reduce.

#define N_CLASSES 21
#define NWAVES    8
#define BLOCK     256
#define CHUNK     128      // elements per wave-chunk: 32 lanes * 4 (16B/lane)
#define PIPE      4
#define MAX_WG    2048
#define STAGE_BYTES (NWAVES * CHUNK * 4)   // 4096 B per stage per array

__device__ __forceinline__ void issue_async_chunk(unsigned ldsO, unsigned ldsG,
                                                  unsigned voff,
                                                  unsigned long long outP,
                                                  unsigned long long gtP) {
  // VDST = LDS byte address (per lane), VADDR = 32-bit byte offset, SADDR = base.
  asm volatile("global_load_async_to_lds_b128 %0, %1, %2 th:TH_LOAD_NT"
               :: "v"(ldsO), "v"(voff), "s"(outP) : "memory");
  asm volatile("global_load_async_to_lds_b128 %0, %1, %2 th:TH_LOAD_NT"
               :: "v"(ldsG), "v"(voff), "s"(gtP) : "memory");
}

__global__ void __launch_bounds__(BLOCK)
mse_partial_kernel(const float* __restrict__ outp,
                   const int* __restrict__ gt,
                   float* __restrict__ ws,
                   unsigned numChunks, unsigned waveStride) {
  __shared__ __align__(16) float sOut[PIPE][NWAVES][CHUNK];  // 16 KB
  __shared__ __align__(16) int   sGt [PIPE][NWAVES][CHUNK];  // 16 KB
  // bins[wave][class][lane]: addr/4 = (w*21+c)*32 + lane -> bank-conflict-free
  __shared__ float binSq [NWAVES][N_CLASSES][32];            // 21 KB
  __shared__ float binCnt[NWAVES][N_CLASSES][32];            // 21 KB
  __shared__ float red[2 * N_CLASSES * 32];                  // 5.25 KB

  const int tid  = threadIdx.x;
  const int w    = tid >> 5;
  const int lane = tid & 31;

  for (int i = tid; i < NWAVES * N_CLASSES * 32; i += BLOCK) {
    (&binSq[0][0][0])[i]  = 0.0f;
    (&binCnt[0][0][0])[i] = 0.0f;
  }
  __syncthreads();

  const unsigned long long outP = (unsigned long long)outp;
  const unsigned long long gtP  = (unsigned long long)gt;
  const unsigned laneByte = (unsigned)lane << 4;
  const unsigned ldsObase = (unsigned)(size_t)&sOut[0][w][0] + laneByte;
  const unsigned ldsGbase = (unsigned)(size_t)&sGt [0][w][0] + laneByte;

  const unsigned c0 = (unsigned)blockIdx.x * NWAVES + (unsigned)w;  // uniform
  const unsigned S  = waveStride;

  // Prologue: issue up to 3 stages ahead.
  for (unsigned k = 0; k < PIPE - 1; ++k) {
    const unsigned ck = c0 + k * S;
    if (ck < numChunks)
      issue_async_chunk(ldsObase + k * STAGE_BYTES, ldsGbase + k * STAGE_BYTES,
                        (ck << 9) + laneByte, outP, gtP);
  }

  unsigned stage = 0;
  for (unsigned cur = c0; cur < numChunks; cur += S) {
    const unsigned rem = numChunks - cur;   // >= 1
    if (rem > 3 * S) {
      const unsigned st3 = (stage + 3) & 3;
      issue_async_chunk(ldsObase + st3 * STAGE_BYTES,
                        ldsGbase + st3 * STAGE_BYTES,
                        ((cur + 3 * S) << 9) + laneByte, outP, gtP);
      asm volatile("s_wait_asynccnt 0x6" ::: "memory");  // oldest stage landed
    } else if (rem > 2 * S) {
      asm volatile("s_wait_asynccnt 0x4" ::: "memory");
    } else if (rem > S) {
      asm volatile("s_wait_asynccnt 0x2" ::: "memory");
    } else {
      asm volatile("s_wait_asynccnt 0x0" ::: "memory");
    }

    const float4 o = *reinterpret_cast<const float4*>(&sOut[stage][w][lane << 2]);
    const int4   g = *reinterpret_cast<const int4  *>(&sGt [stage][w][lane << 2]);

    { int c = g.x; float d = (float)c - o.x;
      atomicAdd(&binSq[w][c][lane], d * d); atomicAdd(&binCnt[w][c][lane], 1.0f); }
    { int c = g.y; float d = (float)c - o.y;
      atomicAdd(&binSq[w][c][lane], d * d); atomicAdd(&binCnt[w][c][lane], 1.0f); }
    { int c = g.z; float d = (float)c - o.z;
      atomicAdd(&binSq[w][c][lane], d * d); atomicAdd(&binCnt[w][c][lane], 1.0f); }
    { int c = g.w; float d = (float)c - o.w;
      atomicAdd(&binSq[w][c][lane], d * d); atomicAdd(&binCnt[w][c][lane], 1.0f); }

    stage = (stage + 1) & 3;
  }
  __syncthreads();

  // Phase 1: all 256 threads fold the 8 waves: red[type*672 + c*32 + l].
  for (int i = tid; i < 2 * N_CLASSES * 32; i += BLOCK) {
    const int type = i / (N_CLASSES * 32);
    const int r    = i - type * (N_CLASSES * 32);
    const float* src = type ? &binCnt[0][0][0] : &binSq[0][0][0];
    float s = 0.0f;
    for (int ww = 0; ww < NWAVES; ++ww)
      s += src[ww * (N_CLASSES * 32) + r];
    red[i] = s;
  }
  __syncthreads();

  // Phase 2: 42 partials per workgroup: [0..20] sq sums, [21..41] counts.
  if (tid < 2 * N_CLASSES) {
    const int base = (tid < N_CLASSES) ? tid * 32
                                       : (N_CLASSES * 32) + (tid - N_CLASSES) * 32;
    float s = 0.0f;
    for (int l = 0; l < 32; ++l) s += red[base + l];
    ws[(size_t)blockIdx.x * (2 * N_CLASSES) + tid] = s;
  }
}

// One workgroup per (class,type) index b in [0,42): tree-reduce ws[:,b] -> ws2[b].
__global__ void __launch_bounds__(BLOCK)
mse_reduce_kernel(const float* __restrict__ ws, float* __restrict__ ws2, int numWG) {
  __shared__ float red[BLOCK];
  const int b = blockIdx.x;
  float s = 0.0f;
  for (int g = threadIdx.x; g < numWG; g += BLOCK)
    s += ws[(size_t)g * (2 * N_CLASSES) + b];
  red[threadIdx.x] = s;
  __syncthreads();
  for (int off = BLOCK / 2; off > 0; off >>= 1) {
    if (threadIdx.x < off) red[threadIdx.x] += red[threadIdx.x + off];
    __syncthreads();
  }
  if (threadIdx.x == 0) ws2[b] = red[0];
}

__global__ void mse_final_kernel(const float* __restrict__ ws2,
                                 const float* __restrict__ outp,
                                 const int* __restrict__ gt,
                                 unsigned tailStart, unsigned n,
                                 float* __restrict__ out) {
  const int t = threadIdx.x;
  if (t >= N_CLASSES) return;
  float sq  = ws2[t];
  float cnt = ws2[N_CLASSES + t];
  for (unsigned i = tailStart; i < n; ++i) {   // remainder (empty for 2^25)
    const int c = gt[i];
    if (c == t) { float d = (float)c - outp[i]; sq += d * d; cnt += 1.0f; }
  }
  out[t] = sq / fmaxf(cnt, 1e-5f);
}

extern "C" void kernel_launch(void* const* d_in, const int* in_sizes, int n_in,
                              void* d_out, int out_size, void* d_ws, size_t ws_size,
                              hipStream_t stream) {
  (void)n_in; (void)out_size;
  const float* outp = (const float*)d_in[0];
  const int*   gt   = (const int*)d_in[1];
  float*       ws   = (float*)d_ws;
  float*       out  = (float*)d_out;

  const unsigned n         = (unsigned)in_sizes[0];
  const unsigned numChunks = n / CHUNK;

  // ws layout: [G][42] partials, then [42] reduced.
  size_t maxG = (ws_size / sizeof(float) >= 2 * N_CLASSES)
                    ? (ws_size / sizeof(float) - 2 * N_CLASSES) / (2 * N_CLASSES)
                    : 1;
  int G = (int)(maxG < (size_t)MAX_WG ? maxG : (size_t)MAX_WG);
  if (G < 1) G = 1;
  float* ws2 = ws + (size_t)G * (2 * N_CLASSES);
  const unsigned waveStride = (unsigned)G * NWAVES;

  mse_partial_kernel<<<dim3(G), dim3(BLOCK), 0, stream>>>(outp, gt, ws,
                                                          numChunks, waveStride);
  mse_reduce_kernel<<<dim3(2 * N_CLASSES), dim3(BLOCK), 0, stream>>>(ws, ws2, G);
  mse_final_kernel<<<dim3(1), dim3(32), 0, stream>>>(ws2, outp, gt,
                                                     numChunks * CHUNK, n, out);
}